// SOMModule_32452772889035
// MI455X (gfx1250) — compile-verified
//
#include <hip/hip_runtime.h>
#include <stdint.h>

typedef __attribute__((ext_vector_type(2))) float v2f;
typedef __attribute__((ext_vector_type(8))) float v8f;

// The async-LDS builtin takes pointers to 16-byte int vectors:
// (AS1 v4i* global_src, AS3 v4i* lds_dst, imm offset, imm cpol)
typedef int v4i __attribute__((vector_size(16)));
typedef __attribute__((address_space(1))) v4i as1_v4i;
typedef __attribute__((address_space(3))) v4i as3_v4i;

// Problem constants
static constexpr int Lc   = 128;  // L
static constexpr int Dc   = 768;  // D
static constexpr int KC   = 24;   // D-chunk staged in LDS (768/24 = 32 chunks)
static constexpr int LDSS = 28;   // padded LDS row stride (floats): 112B rows -> 16B aligned, conflict-free
static constexpr int BUFSZ = Lc * LDSS;           // floats per buffer
static constexpr int NCHUNK = Dc / KC;            // 32

// ---- async global -> LDS (ASYNCcnt path), b128 per call ----
static __device__ __forceinline__ void async_ld128(const float* g, float* l) {
    unsigned int       loff  = (unsigned int)(uintptr_t)l;   // LDS offset lives in low 32 bits
    unsigned long long gaddr = (unsigned long long)(uintptr_t)g;
#if __has_builtin(__builtin_amdgcn_global_load_async_to_lds_b128)
    __builtin_amdgcn_global_load_async_to_lds_b128(
        (as1_v4i*)gaddr, (as3_v4i*)loff, 0, 0);
#else
    asm volatile("global_load_async_to_lds_b128 %0, %1, off"
                 :: "v"(loff), "v"(gaddr) : "memory");
#endif
}

static __device__ __forceinline__ void wait_async_le6() {
#if __has_builtin(__builtin_amdgcn_s_wait_asynccnt)
    __builtin_amdgcn_s_wait_asynccnt(6);
#else
    asm volatile("s_wait_asynccnt 0x6" ::: "memory");
#endif
}
static __device__ __forceinline__ void wait_async_le0() {
#if __has_builtin(__builtin_amdgcn_s_wait_asynccnt)
    __builtin_amdgcn_s_wait_asynccnt(0);
#else
    asm volatile("s_wait_asynccnt 0x0" ::: "memory");
#endif
}

__global__ __launch_bounds__(256)
void som_simmax_kernel(const float* __restrict__ context, float* __restrict__ out) {
    __shared__ float As[2 * BUFSZ];       // ctx chunks, double buffered (128 x 24, padded)
    __shared__ float Es[2 * BUFSZ];       // ent chunks, double buffered
    __shared__ float partial[Lc][2];      // per-row max per column-half
    __shared__ float wsum[8];

    const int tid   = threadIdx.x;
    const int lane  = tid & 31;
    const int wave  = tid >> 5;
    const int wr    = wave & 3;           // row block 0..3  (32 rows each)
    const int wc    = wave >> 2;          // col block 0..1  (64 cols each)
    const int rowBase = wr * 32;
    const int colBase = wc * 64;
    const int lmod  = lane & 15;          // N / M index within 16
    const int lhalf = lane >> 4;          // A/B: K-pair select; C/D: rows +8
    const int kb    = lhalf * 2;

    const size_t bk = blockIdx.x;                         // 0 .. B*K-1
    const float* ctx = context + bk * (size_t)(2 * Lc * Dc);
    const float* ent = ctx + (size_t)(Lc * Dc);

    // Per-thread async-copy slice: 128 rows x 6 float4 = 768 f4 per matrix, 3 per thread.
    int arow[3], ac4[3];
    #pragma unroll
    for (int i = 0; i < 3; ++i) {
        const int f = tid + i * 256;      // 0..767
        arow[i] = f / 6;
        ac4[i]  = (f % 6) * 4;            // 0,4,8,12,16,20
    }

    v8f acc[2][4];
    #pragma unroll
    for (int i = 0; i < 2; ++i)
        #pragma unroll
        for (int j = 0; j < 4; ++j)
            acc[i][j] = (v8f){0.f, 0.f, 0.f, 0.f, 0.f, 0.f, 0.f, 0.f};

    // ---- prologue: async-issue chunks 0 and 1 ----
    #pragma unroll
    for (int c = 0; c < 2; ++c) {
        float* Ab = &As[c * BUFSZ];
        float* Eb = &Es[c * BUFSZ];
        const int d0 = c * KC;
        #pragma unroll
        for (int i = 0; i < 3; ++i) {
            async_ld128(ctx + arow[i] * Dc + d0 + ac4[i], &Ab[arow[i] * LDSS + ac4[i]]);
            async_ld128(ent + arow[i] * Dc + d0 + ac4[i], &Eb[arow[i] * LDSS + ac4[i]]);
        }
    }

    for (int c = 0; c < NCHUNK; ++c) {
        // wait for chunk c to land in LDS (<=6 leaves chunk c+1 in flight)
        if (c + 1 < NCHUNK) wait_async_le6(); else wait_async_le0();
        __syncthreads();

        const int buf = c & 1;
        const float* Ab = &As[buf * BUFSZ];
        const float* Eb = &Es[buf * BUFSZ];

        // ---- 16x16x4 f32 WMMA over the chunk ----
        #pragma unroll
        for (int kk = 0; kk < KC; kk += 4) {
            const int ko = kk + kb;
            v2f a0 = *(const v2f*)&Ab[(rowBase +      lmod) * LDSS + ko];
            v2f a1 = *(const v2f*)&Ab[(rowBase + 16 + lmod) * LDSS + ko];
            v2f b0 = *(const v2f*)&Eb[(colBase +      lmod) * LDSS + ko];
            v2f b1 = *(const v2f*)&Eb[(colBase + 16 + lmod) * LDSS + ko];
            v2f b2 = *(const v2f*)&Eb[(colBase + 32 + lmod) * LDSS + ko];
            v2f b3 = *(const v2f*)&Eb[(colBase + 48 + lmod) * LDSS + ko];

            acc[0][0] = __builtin_amdgcn_wmma_f32_16x16x4_f32(false, a0, false, b0, (short)0, acc[0][0], false, false);
            acc[0][1] = __builtin_amdgcn_wmma_f32_16x16x4_f32(false, a0, false, b1, (short)0, acc[0][1], false, false);
            acc[0][2] = __builtin_amdgcn_wmma_f32_16x16x4_f32(false, a0, false, b2, (short)0, acc[0][2], false, false);
            acc[0][3] = __builtin_amdgcn_wmma_f32_16x16x4_f32(false, a0, false, b3, (short)0, acc[0][3], false, false);
            acc[1][0] = __builtin_amdgcn_wmma_f32_16x16x4_f32(false, a1, false, b0, (short)0, acc[1][0], false, false);
            acc[1][1] = __builtin_amdgcn_wmma_f32_16x16x4_f32(false, a1, false, b1, (short)0, acc[1][1], false, false);
            acc[1][2] = __builtin_amdgcn_wmma_f32_16x16x4_f32(false, a1, false, b2, (short)0, acc[1][2], false, false);
            acc[1][3] = __builtin_amdgcn_wmma_f32_16x16x4_f32(false, a1, false, b3, (short)0, acc[1][3], false, false);
        }
        __syncthreads();

        // ---- buffer `buf` is free: async-issue chunk c+2 into it ----
        if (c + 2 < NCHUNK) {
            float* An = &As[buf * BUFSZ];
            float* En = &Es[buf * BUFSZ];
            const int d0 = (c + 2) * KC;
            #pragma unroll
            for (int i = 0; i < 3; ++i) {
                async_ld128(ctx + arow[i] * Dc + d0 + ac4[i], &An[arow[i] * LDSS + ac4[i]]);
                async_ld128(ent + arow[i] * Dc + d0 + ac4[i], &En[arow[i] * LDSS + ac4[i]]);
            }
        }
    }

    // ---- per-row max over this wave's 64 columns ----
    // C/D layout: VGPR r of lanes 0-15 -> row (base + i*16 + r), col = lane;
    //             lanes 16-31 -> row (base + i*16 + r + 8), col = lane-16.
    #pragma unroll
    for (int i = 0; i < 2; ++i) {
        #pragma unroll
        for (int r = 0; r < 8; ++r) {
            float m = acc[i][0][r];
            m = fmaxf(m, acc[i][1][r]);
            m = fmaxf(m, acc[i][2][r]);
            m = fmaxf(m, acc[i][3][r]);
            m = fmaxf(m, __shfl_xor(m, 1, 16));
            m = fmaxf(m, __shfl_xor(m, 2, 16));
            m = fmaxf(m, __shfl_xor(m, 4, 16));
            m = fmaxf(m, __shfl_xor(m, 8, 16));
            if (lmod == 0) {
                const int row = rowBase + i * 16 + r + lhalf * 8;
                partial[row][wc] = m;
            }
        }
    }
    __syncthreads();

    // ---- combine column halves, sum row-maxima over 128 rows ----
    float v = 0.f;
    if (tid < Lc) v = fmaxf(partial[tid][0], partial[tid][1]);
    v += __shfl_xor(v, 16, 32);
    v += __shfl_xor(v, 8, 32);
    v += __shfl_xor(v, 4, 32);
    v += __shfl_xor(v, 2, 32);
    v += __shfl_xor(v, 1, 32);
    if (lane == 0) wsum[wave] = v;
    __syncthreads();
    if (tid == 0) {
        float s = 0.f;
        #pragma unroll
        for (int w = 0; w < 8; ++w) s += wsum[w];
        out[bk] = s;
    }
}

extern "C" void kernel_launch(void* const* d_in, const int* in_sizes, int n_in,
                              void* d_out, int out_size, void* d_ws, size_t ws_size,
                              hipStream_t stream) {
    const float* context = (const float*)d_in[0];
    float* out = (float*)d_out;
    // B*K = 16*64 = 1024 independent (L x L x D) problems, one block each.
    som_simmax_kernel<<<dim3(1024), dim3(256), 0, stream>>>(context, out);
}